// WavefunctionEmbedding_28278064676910
// MI455X (gfx1250) — compile-verified
//
#include <hip/hip_runtime.h>

// WavefunctionEmbedding for MI455X (gfx1250, wave32).
// d_in[0]: coords  [B,N,3] f32
// d_in[1]: key_padding_mask [B,N] bool (1 byte each; True = padded)
// d_in[2]: wavenumbers [K] f32
// d_out  : [B,N,2K] f32   (re/im interleaved per wavenumber)

typedef float v2f __attribute__((ext_vector_type(2)));
typedef float v8f __attribute__((ext_vector_type(8)));

#define BATCH   2
#define NPTS    1536
#define KHALF   256
#define TI      16                 // target rows per workgroup
#define JCHUNK  512                // source columns per LDS chunk
#define JS      (JCHUNK + 4)       // padded LDS row stride (dwords) -> conflict-free
#define NCHUNKS (NPTS / JCHUNK)    // 3
#define INV4PI  0.07957747154594767f

__global__ __launch_bounds__(256) void wavefn_kernel(
    const float* __restrict__ coords,
    const unsigned char* __restrict__ pad,
    const float* __restrict__ wn,
    float* __restrict__ out)
{
    __shared__ float sD2[TI * JS];     // ~33 KB: squared distances for current chunk
    __shared__ float sMask[NPTS];      // 6 KB: 1.0 = valid source, 0.0 = padded
    __shared__ float sNorm[TI];        // |c_i|^2 for the 16 target rows

    const int tid  = threadIdx.x;
    const int wave = tid >> 5;
    const int lane = tid & 31;
    const int b     = blockIdx.x / (NPTS / TI);
    const int ibase = (blockIdx.x % (NPTS / TI)) * TI;

    const float* cb = coords + (size_t)b * NPTS * 3;

    // ---- preload mask + target row norms into LDS ----
    for (int idx = tid; idx < NPTS; idx += 256)
        sMask[idx] = pad[(size_t)b * NPTS + idx] ? 0.0f : 1.0f;
    if (tid < TI) {
        const float* ci = cb + (size_t)(ibase + tid) * 3;
        float x = ci[0], y = ci[1], z = ci[2];
        sNorm[tid] = x * x + y * y + z * z;
    }

    // ---- WMMA A-fragment: 16x4 f32, A row M = (-2x_i, -2y_i, -2z_i, 1) ----
    // ISA layout (32-bit A 16x4): lane<16 -> {K0,K1} of row M=lane;
    //                             lane>=16 -> {K2,K3} of row M=lane-16.
    v2f afrag;
    {
        const float* ci = cb + (size_t)(ibase + (lane & 15)) * 3;
        float x = ci[0], y = ci[1], z = ci[2];
        afrag.x = (lane < 16) ? -2.0f * x : -2.0f * z;
        afrag.y = (lane < 16) ? -2.0f * y : 1.0f;
    }

    // ---- per-lane wavenumbers: m = lane + 32*t ----
    float kv[8];
    #pragma unroll
    for (int t = 0; t < 8; ++t)
        kv[t] = wn[lane + 32 * t];

    const int i0 = ibase + 2 * wave;
    const int i1 = i0 + 1;

    float re0[8], im0[8], re1[8], im1[8];
    #pragma unroll
    for (int t = 0; t < 8; ++t) { re0[t] = 0.f; im0[t] = 0.f; re1[t] = 0.f; im1[t] = 0.f; }

    for (int c = 0; c < NCHUNKS; ++c) {
        __syncthreads();   // previous chunk's phase-2 reads finished; LDS preload visible
        const int chunkbase = c * JCHUNK;

        // ================= Phase 1: d^2 tile via v_wmma_f32_16x16x4_f32 =====
        // wave w handles 4 tiles of 16 columns: covers [w*64, w*64+64) of chunk
        #pragma unroll
        for (int tt = 0; tt < 4; ++tt) {
            const int jl0 = (wave * 4 + tt) * 16;            // column base within chunk
            const int jg  = chunkbase + jl0 + (lane & 15);   // global source index
            const float* cj = cb + (size_t)jg * 3;
            float x = cj[0], y = cj[1], z = cj[2];
            float nj = x * x + y * y + z * z;

            // B-fragment 4x16: rows striped across lanes within a VGPR.
            // lane<16 -> {row K0 (=x_j), row K2 (=z_j)} at N=lane
            // lane>=16-> {row K1 (=y_j), row K3 (=|c_j|^2)} at N=lane-16
            v2f bfrag;
            bfrag.x = (lane < 16) ? x : y;
            bfrag.y = (lane < 16) ? z : nj;

            v8f acc = {};
            acc = __builtin_amdgcn_wmma_f32_16x16x4_f32(
                false, afrag, false, bfrag, (short)0, acc, false, false);

            // C layout: lane<16 holds rows 0..7 (col=lane); lane>=16 rows 8..15.
            const int rbase = (lane < 16) ? 0 : 8;
            const int colp  = jl0 + (lane & 15);
            #pragma unroll
            for (int v = 0; v < 8; ++v) {
                float d2 = acc[v] + sNorm[rbase + v];
                d2 = fmaxf(d2, 0.0f) + 1e-12f;
                sD2[(rbase + v) * JS + colp] = d2;
            }
        }
        __syncthreads();

        // ================= Phase 2: sincos accumulation =====================
        for (int jl = 0; jl < JCHUNK; ++jl) {
            const int j  = chunkbase + jl;
            const float mj = sMask[j];

            float d2a = sD2[(2 * wave + 0) * JS + jl];
            float d2b = sD2[(2 * wave + 1) * JS + jl];
            float inva = __builtin_amdgcn_rsqf(d2a);    // v_rsq_f32: 1/r
            float invb = __builtin_amdgcn_rsqf(d2b);
            float ra = d2a * inva;                      // r = d2 * rsqrt(d2)
            float rb = d2b * invb;
            float ampa = (j == i0) ? 0.0f : INV4PI * mj * inva;
            float ampb = (j == i1) ? 0.0f : INV4PI * mj * invb;

            #pragma unroll
            for (int t = 0; t < 8; ++t) {
                float pa = kv[t] * ra;
                float pb = kv[t] * rb;
                re0[t] = __builtin_fmaf(__cosf(pa), ampa, re0[t]);
                im0[t] = __builtin_fmaf(__sinf(pa), ampa, im0[t]);
                re1[t] = __builtin_fmaf(__cosf(pb), ampb, re1[t]);
                im1[t] = __builtin_fmaf(__sinf(pb), ampb, im1[t]);
            }
        }
        __syncthreads();
    }

    // ---- epilogue: packed (re,im) stores, contiguous across lanes ----
    float* o0 = out + (size_t)(b * NPTS + i0) * (2 * KHALF);
    float* o1 = out + (size_t)(b * NPTS + i1) * (2 * KHALF);
    #pragma unroll
    for (int t = 0; t < 8; ++t) {
        const int m = lane + 32 * t;
        float2 va; va.x = re0[t]; va.y = im0[t];
        float2 vb; vb.x = re1[t]; vb.y = im1[t];
        *(float2*)(o0 + 2 * m) = va;
        *(float2*)(o1 + 2 * m) = vb;
    }
}

extern "C" void kernel_launch(void* const* d_in, const int* in_sizes, int n_in,
                              void* d_out, int out_size, void* d_ws, size_t ws_size,
                              hipStream_t stream) {
    (void)in_sizes; (void)n_in; (void)d_ws; (void)ws_size; (void)out_size;
    const float*         coords = (const float*)d_in[0];
    const unsigned char* pad    = (const unsigned char*)d_in[1];
    const float*         wn     = (const float*)d_in[2];
    float*               out    = (float*)d_out;

    dim3 grid(BATCH * (NPTS / TI));   // 192 workgroups
    dim3 block(256);                  // 8 wave32 waves
    wavefn_kernel<<<grid, block, 0, stream>>>(coords, pad, wn, out);
}